// SocialCircleLayer_73048803770821
// MI455X (gfx1250) — compile-verified
//
#include <hip/hip_runtime.h>

typedef __attribute__((ext_vector_type(4))) float f4;
typedef __attribute__((ext_vector_type(2))) float v2f;
typedef __attribute__((ext_vector_type(8))) float v8f;

#define NBATCH   16384
#define NNEI     128
#define NP       8
#define TWO_PI_F 6.28318530717958647692f
#define INV_BIN  1.27323954473516268615f   /* 1 / (2*pi/8) = 4/pi */
#define EPS_F    1.0e-4f

// One wave (32 lanes) per batch; 8 waves (256 threads) per block.
// Phase 1: stream 32 floats/neighbor (coalesced 128-bit NT loads), compute
//          vel/dist/dir/mask, stash features column-major in LDS, write f_direction.
// Phase 2: sums(16x16) = onehot^T (16x128) x [feats|1] (128x16) via 32 steps of
//          v_wmma_f32_16x16x4_f32; column N=3 of B is the constant 1.0 so
//          D[p][3] = count[p]. All loads in the loop are EXEC-uniform.
__global__ __launch_bounds__(256) void social_circle_kernel(
    const float* __restrict__ nei,
    float* __restrict__ out_social,   // (16384, 8, 3)
    float* __restrict__ out_dir)      // (16384, 128)
{
    __shared__ int   s_idx[8][NNEI];          // angle bin per neighbor (-1 = masked)
    __shared__ float s_feat[8][3][NNEI];      // column-major: [feature][neighbor]

    const int lane = threadIdx.x & 31;
    const int wave = threadIdx.x >> 5;
    const long b   = (long)blockIdx.x * 8 + wave;

    // ---------------- Phase 1: streaming feature extraction ----------------
    for (int c = 0; c < 4; ++c) {
        const int n = c * 32 + lane;
        const f4* p = (const f4*)(nei + ((long)b * NNEI + n) * 32);

        f4 q0 = __builtin_nontemporal_load(p + 0);
        f4 q1 = __builtin_nontemporal_load(p + 1);
        f4 q2 = __builtin_nontemporal_load(p + 2);
        f4 q3 = __builtin_nontemporal_load(p + 3);
        f4 q4 = __builtin_nontemporal_load(p + 4);
        f4 q5 = __builtin_nontemporal_load(p + 5);
        f4 q6 = __builtin_nontemporal_load(p + 6);
        f4 q7 = __builtin_nontemporal_load(p + 7);

        // mask: sum of all 32 elements != 0
        f4 s01 = q0 + q1, s23 = q2 + q3, s45 = q4 + q5, s67 = q6 + q7;
        f4 s = (s01 + s23) + (s45 + s67);
        float total = (s.x + s.y) + (s.z + s.w);

        // first point = q0.xy (t=0), last point = q7.zw (t=15)
        float px = q7.z, py = q7.w;
        float dx = px - q0.x, dy = py - q0.y;

        float vel  = sqrtf(dx * dx + dy * dy);
        float dist = sqrtf(px * px + py * py);
        float dir  = atan2f(py, px);
        if (dir < 0.0f) dir += TWO_PI_F;     // mod 2*pi of atan2 range

        int idx = (int)(dir * INV_BIN);      // dir >= 0, trunc == floor
        if (total == 0.0f) idx = -1;         // masked neighbor

        s_idx[wave][n]     = idx;
        s_feat[wave][0][n] = vel;
        s_feat[wave][1][n] = dist;
        s_feat[wave][2][n] = dir;

        __builtin_nontemporal_store(dir, &out_dir[b * NNEI + n]);
    }

    __syncthreads();

    // ---------------- Phase 2: WMMA bin-reduction -------------------------
    // A (16x4, M=bin, K=neighbor chunk): VGPR0 = {K0 lanes0-15, K2 lanes16-31},
    //                                    VGPR1 = {K1, K3}
    // B (4x16, K=neighbor, N=feature):   VGPR0 = {K0 row, K2 row}, VGPR1 = {K1, K3}
    const int  m      = lane & 15;       // bin index for A, feature column for B
    const int  hi     = lane >> 4;       // which K pair this lane half holds
    const int  col    = (m < 3) ? m : 0; // clamped (always-valid) feature column
    const bool isFeat = (m < 3);
    const float ext   = (m == 3) ? 1.0f : 0.0f;   // count column / zero padding

    const int*   ip = &s_idx[wave][hi ? 2 : 0];
    const float* fp = &s_feat[wave][col][hi ? 2 : 0];

    v8f acc = {};
#pragma unroll 4
    for (int k = 0; k < 32; ++k) {
        // idx pair {ka, ka+1}: adjacent ints -> one ds_load_b64
        const int ia = ip[0];
        const int ib = ip[1];
        // feature pair {feats[ka][col], feats[ka+1][col]}: adjacent -> ds_load_b64
        const float fx = fp[0];
        const float fy = fp[1];

        v2f amat, bmat;
        amat.x = (ia == m) ? 1.0f : 0.0f;
        amat.y = (ib == m) ? 1.0f : 0.0f;
        bmat.x = isFeat ? fx : ext;
        bmat.y = isFeat ? fy : ext;

        acc = __builtin_amdgcn_wmma_f32_16x16x4_f32(
                  false, amat, false, bmat, (short)0, acc, false, false);

        ip += 4;
        fp += 4;
    }

    // D layout: VGPR j, lanes 0-15 -> M=j, N=lane. Lanes 0..2 hold sums[j][f],
    // lane 3 holds count[j]. Broadcast lane 3 via ds_bpermute, normalize, store.
#pragma unroll
    for (int j = 0; j < 8; ++j) {
        float sj  = acc[j];
        float cnt = __int_as_float(
            __builtin_amdgcn_ds_bpermute(3 << 2, __float_as_int(sj)));
        float val = sj / (cnt + EPS_F);
        if (lane < 3) {
            out_social[b * (NP * 3) + j * 3 + lane] = val;
        }
    }
}

extern "C" void kernel_launch(void* const* d_in, const int* in_sizes, int n_in,
                              void* d_out, int out_size, void* d_ws, size_t ws_size,
                              hipStream_t stream) {
    // inputs: d_in[0] = trajs (unused by the math), d_in[1] = nei_trajs
    const float* nei = (const float*)d_in[1];
    float* out       = (float*)d_out;

    float* out_social = out;                         // 16384*8*3 floats
    float* out_dir    = out + (long)NBATCH * NP * 3; // 16384*128 floats

    dim3 grid(NBATCH / 8);
    dim3 block(256);
    hipLaunchKernelGGL(social_circle_kernel, grid, block, 0, stream,
                       nei, out_social, out_dir);
}